// grid_crop_23648089932317
// MI455X (gfx1250) — compile-verified
//
#include <hip/hip_runtime.h>

// 128-bit native vector types (lower to global_load_b128 / global_store_b128,
// and are accepted by the nontemporal builtins, unlike HIP's struct float4).
typedef float        f4 __attribute__((ext_vector_type(4)));
typedef unsigned int u4 __attribute__((ext_vector_type(4)));

// image: [B=64, C=3, H=512, W=512] fp32; mask: [64, 64] bool (1 byte each).
// out[b,c,h,w] = mask[h/8, w/8] ? -1.0f : image[b,c,h,w]
//
// Pure streaming kernel:
//  - one f4 (16B) chunk per thread per iteration, grid-stride persistent blocks
//  - 4KB mask cached in LDS once per block (reused ~24x per thread)
//  - nontemporal load/store: 400MB stream has zero reuse, keep it out of L2's way
//  - masked chunks skip the image load entirely (rate=0.5 -> ~halves read BW)
//  - global_prefetch_b8 one stride ahead to help hide HBM latency
__global__ __launch_bounds__(256) void masked_blockfill_kernel(
    const f4* __restrict__ img,
    const unsigned char* __restrict__ mask,
    f4* __restrict__ out,
    long long n4)
{
    __shared__ __align__(16) unsigned char smask[64 * 64];  // 4 KB

    // Cooperative mask load: 256 threads x 16 B = 4096 B. blockDim.x must be 256.
    ((u4*)smask)[threadIdx.x] = ((const u4*)mask)[threadIdx.x];
    __syncthreads();

    const f4 fill = { -1.0f, -1.0f, -1.0f, -1.0f };

    long long i = (long long)blockIdx.x * 256 + threadIdx.x;
    const long long stride = (long long)gridDim.x * 256;

    for (; i < n4; i += stride) {
        // Speculative prefetch of next iteration's image chunk (global_prefetch_b8).
        if (i + stride < n4)
            __builtin_prefetch(&img[i + stride], 0, 1);

        // W = 512 -> 128 f4-chunks per row; chunk never straddles an 8-wide block.
        const int w4 = (int)((unsigned)(i      ) & 127u);  // chunk col in [0,128)
        const int h  = (int)((i >> 7) & 511);              // row in [0,512)
        const int mi = ((h >> 3) << 6) | (w4 >> 1);        // (h/8)*64 + (w4*4)/8

        if (smask[mi]) {
            // Masked block: constant fill, no image read needed.
            __builtin_nontemporal_store(fill, &out[i]);
        } else {
            f4 v = __builtin_nontemporal_load(&img[i]);
            __builtin_nontemporal_store(v, &out[i]);
        }
    }
}

extern "C" void kernel_launch(void* const* d_in, const int* in_sizes, int n_in,
                              void* d_out, int out_size, void* d_ws, size_t ws_size,
                              hipStream_t stream)
{
    const f4*            img  = (const f4*)d_in[0];             // fp32 image
    const unsigned char* mask = (const unsigned char*)d_in[1];  // bool mask, 1B/elem
    f4*                  outp = (f4*)d_out;

    const long long n4 = (long long)out_size / 4;  // 12,582,912 f4 chunks

    // Persistent grid: 2048 blocks x 8 wave32 waves; each thread runs ~24
    // iterations so the per-block 4KB mask->LDS fill amortizes away.
    const int blocks  = 2048;
    const int threads = 256;
    masked_blockfill_kernel<<<blocks, threads, 0, stream>>>(img, mask, outp, n4);
}